// LiteMLA3d_8151847928361
// MI455X (gfx1250) — compile-verified
//
#include <hip/hip_runtime.h>

typedef __attribute__((ext_vector_type(2))) float v2f;
typedef __attribute__((ext_vector_type(8))) float v8f;

constexpr int kN = 16384;   // D*H*W = 16*32*32

// ---- CDNA5 async global->LDS copy (ASYNCcnt-tracked, see CDNA5 ISA) --
__device__ __forceinline__ void async_ld_b128(uint32_t lds_off, const void* g) {
  asm volatile("global_load_async_to_lds_b128 %0, %1, off"
               :: "v"(lds_off), "v"(g) : "memory");
}
__device__ __forceinline__ void async_wait0() {
  asm volatile("s_wait_asynccnt 0" ::: "memory");
}
__device__ __forceinline__ uint32_t lds_addr(const void* p) {
  return (uint32_t)(uintptr_t)p;   // low 32 bits of generic addr = LDS offset
}

// cat = concat([qkv(768ch), agg2(768ch)]); heads never straddle the boundary.
__device__ __forceinline__ const float* cat_row(const float* __restrict__ qkv,
                                                const float* __restrict__ agg2,
                                                int b, int c) {
  const float* base = (c < 768) ? qkv : agg2;
  const int cc = (c < 768) ? c : (c - 768);
  return base + ((size_t)b * 768 + cc) * (size_t)kN;
}

#define WMMA_F32(A, B, C) \
  __builtin_amdgcn_wmma_f32_16x16x4_f32(false, (A), false, (B), (short)0, (C), false, false)

// ---------------- qkv 1x1 conv: y[b,m,n] = sum_k w[m,k] * x[b,k,n] ----------
// 4 M-tiles per wave: B fragments reused 4x -> 4x less activation traffic.
__global__ __launch_bounds__(256) void k_qkv(const float* __restrict__ x,
                                             const float* __restrict__ w,
                                             float* __restrict__ y) {
  __shared__ __align__(16) float lw[64 * 256];   // 64 KB weight panel
  const int tid = threadIdx.x;
  const int m0  = blockIdx.y * 64;
  const int b   = blockIdx.z;
  {
    const char* wsrc = (const char*)(w + (size_t)m0 * 256);
    const uint32_t lb = lds_addr(&lw[0]);
    for (int i = tid; i < 4096; i += 256)               // 4096 x 16B
      async_ld_b128(lb + (uint32_t)i * 16u, wsrc + (size_t)i * 16);
    async_wait0();
  }
  __syncthreads();
  __builtin_amdgcn_s_wait_tensorcnt(0);
  const int wave = tid >> 5, lane = tid & 31;
  const int col  = blockIdx.x * 128 + wave * 16 + (lane & 15);
  const int koff = (lane < 16) ? 0 : 2;
  const int lm   = lane & 15;
  const float* xb = x + (size_t)b * 256 * kN;
  v8f c0 = {}, c1 = {}, c2 = {}, c3 = {};
#pragma unroll 2
  for (int k = 0; k < 256; k += 4) {
    const int kk = k + koff;
    v2f Bv;
    Bv.x = xb[(size_t)kk * kN + col];
    Bv.y = xb[(size_t)(kk + 1) * kN + col];
    v2f A;
    A.x = lw[(0 * 16 + lm) * 256 + kk]; A.y = lw[(0 * 16 + lm) * 256 + kk + 1];
    c0 = WMMA_F32(A, Bv, c0);
    A.x = lw[(1 * 16 + lm) * 256 + kk]; A.y = lw[(1 * 16 + lm) * 256 + kk + 1];
    c1 = WMMA_F32(A, Bv, c1);
    A.x = lw[(2 * 16 + lm) * 256 + kk]; A.y = lw[(2 * 16 + lm) * 256 + kk + 1];
    c2 = WMMA_F32(A, Bv, c2);
    A.x = lw[(3 * 16 + lm) * 256 + kk]; A.y = lw[(3 * 16 + lm) * 256 + kk + 1];
    c3 = WMMA_F32(A, Bv, c3);
  }
  const int rbase = (lane >> 4) * 8;
  float* yb = y + ((size_t)b * 768 + m0) * kN;
#pragma unroll
  for (int r = 0; r < 8; ++r) {
    yb[(size_t)(r + rbase) * kN + col]      = c0[r];
    yb[(size_t)(16 + r + rbase) * kN + col] = c1[r];
    yb[(size_t)(32 + r + rbase) * kN + col] = c2[r];
    yb[(size_t)(48 + r + rbase) * kN + col] = c3[r];
  }
}

// ---------------- depthwise 5x5x5 conv, pad 2 ------------------------------
// z-slab (8 out planes + halo = 12 planes, 48 KB) staged in LDS via async DMA.
__global__ __launch_bounds__(256) void k_dw(const float* __restrict__ qkv,
                                            const float* __restrict__ wdw,
                                            float* __restrict__ agg) {
  __shared__ __align__(16) float tile[12 * 1024];
  __shared__ float wsm[125];
  const int tid = threadIdx.x;
  const int zo  = blockIdx.x * 8;          // output z base (0 or 8)
  const int c   = blockIdx.y, b = blockIdx.z;
  const float* src = qkv + ((size_t)b * 768 + c) * kN;
  if (tid < 125) wsm[tid] = wdw[(size_t)c * 125 + tid];
  const uint32_t tb = lds_addr(&tile[0]);
#pragma unroll
  for (int zs = 0; zs < 12; ++zs) {
    const int z = zo - 2 + zs;
    if ((unsigned)z < 16u) {               // 1024 floats = 256 x 16B
      async_ld_b128(tb + (uint32_t)(zs * 4096 + tid * 16),
                    (const char*)(src + (size_t)z * 1024) + (size_t)tid * 16);
    } else {
      float4 zz = {0.f, 0.f, 0.f, 0.f};
      ((float4*)&tile[zs * 1024])[tid] = zz;
    }
  }
  async_wait0();
  __syncthreads();
  for (int it = 0; it < 32; ++it) {
    const int o  = tid + it * 256;         // 0..8191
    const int ox = o & 31, oy = (o >> 5) & 31, oz = o >> 10;   // oz 0..7
    float a = 0.f;
    for (int iz = 0; iz < 5; ++iz) {
      const float* slab = &tile[(oz + iz) * 1024];
      for (int iy = 0; iy < 5; ++iy) {
        const int yy = oy + iy - 2;
        if ((unsigned)yy >= 32u) continue;
        const float* row = slab + yy * 32;
        const float* wr  = &wsm[(iz * 5 + iy) * 5];
#pragma unroll
        for (int ix = 0; ix < 5; ++ix) {
          const int xx = ox + ix - 2;
          if ((unsigned)xx >= 32u) continue;
          a = fmaf(wr[ix], row[xx], a);
        }
      }
    }
    agg[((size_t)b * 768 + c) * kN + (size_t)(zo + oz) * 1024 + oy * 32 + ox] = a;
  }
}

// ---------------- grouped 1x1 conv (24 groups of 32x32), full group/block ---
__global__ __launch_bounds__(256) void k_pw(const float* __restrict__ aggv,
                                            const float* __restrict__ wpw,
                                            float* __restrict__ y) {
  __shared__ __align__(16) float lw[32 * 32];
  const int tid = threadIdx.x;
  const int m0  = blockIdx.y * 32;         // == group * 32
  const int b   = blockIdx.z;
  {
    const uint32_t lb = lds_addr(&lw[0]);  // 1024 floats = 256 x 16B
    async_ld_b128(lb + (uint32_t)tid * 16u,
                  (const char*)(wpw + (size_t)m0 * 32) + (size_t)tid * 16);
    async_wait0();
  }
  __syncthreads();
  const int wave = tid >> 5, lane = tid & 31;
  const int col  = blockIdx.x * 128 + wave * 16 + (lane & 15);
  const int koff = (lane < 16) ? 0 : 2;
  const int lm   = lane & 15;
  const float* xb = aggv + ((size_t)b * 768 + m0) * kN;  // input base = group base
  v8f c0 = {}, c1 = {};
#pragma unroll
  for (int k = 0; k < 32; k += 4) {
    const int kk = k + koff;
    v2f Bv;
    Bv.x = xb[(size_t)kk * kN + col];
    Bv.y = xb[(size_t)(kk + 1) * kN + col];
    v2f A;
    A.x = lw[lm * 32 + kk];        A.y = lw[lm * 32 + kk + 1];
    c0 = WMMA_F32(A, Bv, c0);
    A.x = lw[(16 + lm) * 32 + kk]; A.y = lw[(16 + lm) * 32 + kk + 1];
    c1 = WMMA_F32(A, Bv, c1);
  }
  const int rbase = (lane >> 4) * 8;
  float* yb = y + ((size_t)b * 768 + m0) * kN;
#pragma unroll
  for (int r = 0; r < 8; ++r) {
    yb[(size_t)(r + rbase) * kN + col]      = c0[r];
    yb[(size_t)(16 + r + rbase) * kN + col] = c1[r];
  }
}

// ---------------- vk partial: vk[e,d] += sum_n V[e,n]*relu(K[d,n]) ----------
constexpr int CHK = 128;
__global__ __launch_bounds__(256) void k_vkp(const float* __restrict__ qkv,
                                             const float* __restrict__ agg2,
                                             float* __restrict__ part) {
  __shared__ float stage[65][CHK + 5];   // rows 0..31 = relu(K), 32..63 = V, 64 = ones
  const int tid = threadIdx.x;
  const int ns = blockIdx.x;             // 0..15 n-split
  const int bh = blockIdx.y;             // 0..31 (b*16+h)
  const int b = bh >> 4, h = bh & 15;
  const int cb = h * 96;
  float acc[5] = {0.f, 0.f, 0.f, 0.f, 0.f};
  for (int chn = 0; chn < 8; ++chn) {
    const int nb = ns * 1024 + chn * CHK;
    for (int idx = tid; idx < 65 * CHK; idx += 256) {
      const int r = idx >> 7, nn = idx & (CHK - 1);
      float v;
      if (r < 32)      v = fmaxf(cat_row(qkv, agg2, b, cb + 32 + r)[nb + nn], 0.f);
      else if (r < 64) v = cat_row(qkv, agg2, b, cb + 64 + (r - 32))[nb + nn];
      else             v = 1.0f;
      stage[r][nn] = v;
    }
    __syncthreads();
#pragma unroll
    for (int s = 0; s < 5; ++s) {
      const int p = tid + s * 256;
      if (p < 1056) {
        const int e = p >> 5, dd2 = p & 31;
        float a = acc[s];
        for (int nn = 0; nn < CHK; ++nn)
          a = fmaf(stage[32 + e][nn], stage[dd2][nn], a);
        acc[s] = a;
      }
    }
    __syncthreads();
  }
#pragma unroll
  for (int s = 0; s < 5; ++s) {
    const int p = tid + s * 256;
    if (p < 1056) part[((size_t)bh * 16 + ns) * 1056 + p] = acc[s];
  }
}

__global__ __launch_bounds__(256) void k_vkr(const float* __restrict__ part,
                                             float* __restrict__ vk) {
  const int bh = blockIdx.x, tid = threadIdx.x;
  for (int p = tid; p < 1056; p += 256) {
    float s = 0.f;
    for (int ns = 0; ns < 16; ++ns)
      s += part[((size_t)bh * 16 + ns) * 1056 + p];
    vk[(size_t)bh * 1056 + p] = s;
  }
}

// ---------------- out = (vk @ relu(Q)) rows / (row32 + eps) -----------------
__global__ __launch_bounds__(256) void k_att(const float* __restrict__ qkv,
                                             const float* __restrict__ agg2,
                                             const float* __restrict__ vk,
                                             float* __restrict__ att) {
  __shared__ __align__(16) float lv[1056];   // 33 x 32
  const int tid = threadIdx.x;
  const int h = blockIdx.y, b = blockIdx.z;
  {
    const char* vs = (const char*)(vk + (size_t)(b * 16 + h) * 1056);
    const uint32_t lb = lds_addr(&lv[0]);    // 4224 B = 264 x 16B
    for (int i = tid; i < 264; i += 256)
      async_ld_b128(lb + (uint32_t)i * 16u, vs + (size_t)i * 16);
    async_wait0();
  }
  __syncthreads();
  const int wave = tid >> 5, lane = tid & 31;
  const int col  = blockIdx.x * 128 + wave * 16 + (lane & 15);
  const int koff = (lane < 16) ? 0 : 2;
  const int lm   = lane & 15;
  const int cb   = h * 96;
  v8f a0 = {}, a1 = {}, a2 = {};
#pragma unroll
  for (int k = 0; k < 32; k += 4) {
    const int kk = k + koff;
    v2f Bv;
    Bv.x = fmaxf(cat_row(qkv, agg2, b, cb + kk)[col], 0.f);
    Bv.y = fmaxf(cat_row(qkv, agg2, b, cb + kk + 1)[col], 0.f);
    v2f A0, A1, A2;
    A0.x = lv[lm * 32 + kk];          A0.y = lv[lm * 32 + kk + 1];
    A1.x = lv[(16 + lm) * 32 + kk];   A1.y = lv[(16 + lm) * 32 + kk + 1];
    const bool has2 = (lm == 0);      // only e=32 is real in third M-tile
    A2.x = has2 ? lv[32 * 32 + kk] : 0.f;
    A2.y = has2 ? lv[32 * 32 + kk + 1] : 0.f;
    a0 = WMMA_F32(A0, Bv, a0);
    a1 = WMMA_F32(A1, Bv, a1);
    a2 = WMMA_F32(A2, Bv, a2);
  }
  const float d0  = a2[0];
  const float den = __shfl(d0, lane & 15, 32) + 1e-15f;
  const float inv = 1.0f / den;
  const int rbase = (lane >> 4) * 8;
  float* ab = att + ((size_t)b * 512 + h * 32) * kN;
#pragma unroll
  for (int r = 0; r < 8; ++r) {
    ab[(size_t)(r + rbase) * kN + col]      = a0[r] * inv;
    ab[(size_t)(16 + r + rbase) * kN + col] = a1[r] * inv;
  }
}

// ---------------- proj 1x1 conv (256x512) + fused BatchNorm, 2 M-tiles ------
__global__ __launch_bounds__(256) void k_proj(const float* __restrict__ att,
                                              const float* __restrict__ w,
                                              const float* __restrict__ gamma,
                                              const float* __restrict__ beta,
                                              const float* __restrict__ rmean,
                                              const float* __restrict__ rvar,
                                              float* __restrict__ out) {
  __shared__ __align__(16) float lw[32 * 512];   // 64 KB
  const int tid = threadIdx.x;
  const int m0  = blockIdx.y * 32;
  const int b   = blockIdx.z;
  {
    const char* wsrc = (const char*)(w + (size_t)m0 * 512);
    const uint32_t lb = lds_addr(&lw[0]);        // 4096 x 16B
    for (int i = tid; i < 4096; i += 256)
      async_ld_b128(lb + (uint32_t)i * 16u, wsrc + (size_t)i * 16);
    async_wait0();
  }
  __syncthreads();
  const int wave = tid >> 5, lane = tid & 31;
  const int col  = blockIdx.x * 128 + wave * 16 + (lane & 15);
  const int koff = (lane < 16) ? 0 : 2;
  const int lm   = lane & 15;
  const float* xb = att + (size_t)b * 512 * kN;
  v8f c0 = {}, c1 = {};
#pragma unroll 2
  for (int k = 0; k < 512; k += 4) {
    const int kk = k + koff;
    v2f Bv;
    Bv.x = xb[(size_t)kk * kN + col];
    Bv.y = xb[(size_t)(kk + 1) * kN + col];
    v2f A;
    A.x = lw[lm * 512 + kk];        A.y = lw[lm * 512 + kk + 1];
    c0 = WMMA_F32(A, Bv, c0);
    A.x = lw[(16 + lm) * 512 + kk]; A.y = lw[(16 + lm) * 512 + kk + 1];
    c1 = WMMA_F32(A, Bv, c1);
  }
  const int rbase = (lane >> 4) * 8;
#pragma unroll
  for (int r = 0; r < 8; ++r) {
    {
      const int row = m0 + r + rbase;
      const float sc = gamma[row] * rsqrtf(rvar[row] + 1e-5f);
      const float bi = beta[row] - rmean[row] * sc;
      out[((size_t)b * 256 + row) * kN + col] = c0[r] * sc + bi;
    }
    {
      const int row = m0 + 16 + r + rbase;
      const float sc = gamma[row] * rsqrtf(rvar[row] + 1e-5f);
      const float bi = beta[row] - rmean[row] * sc;
      out[((size_t)b * 256 + row) * kN + col] = c1[r] * sc + bi;
    }
  }
}

extern "C" void kernel_launch(void* const* d_in, const int* in_sizes, int n_in,
                              void* d_out, int out_size, void* d_ws, size_t ws_size,
                              hipStream_t stream) {
  (void)in_sizes; (void)n_in; (void)out_size; (void)ws_size;
  const float* x     = (const float*)d_in[0];
  const float* wqkv  = (const float*)d_in[1];
  const float* wdw   = (const float*)d_in[2];
  const float* wpw   = (const float*)d_in[3];
  const float* wproj = (const float*)d_in[4];
  const float* gamma = (const float*)d_in[5];
  const float* beta  = (const float*)d_in[6];
  const float* rmean = (const float*)d_in[7];
  const float* rvar  = (const float*)d_in[8];
  float* out = (float*)d_out;

  char* ws = (char*)d_ws;
  const size_t szQ = (size_t)2 * 768 * kN * sizeof(float);   // 100.66 MB
  float* qkv  = (float*)(ws);
  float* agg  = (float*)(ws + szQ);
  float* agg2 = (float*)(ws + 2 * szQ);
  float* att  = agg;   // agg is dead after k_pw; reuse for attention output
  float* vkp  = (float*)(ws + 3 * szQ);
  float* vk   = (float*)(ws + 3 * szQ + (size_t)32 * 16 * 1056 * sizeof(float));

  const dim3 blk(256);
  k_qkv<<<dim3(128, 12, 2),  blk, 0, stream>>>(x, wqkv, qkv);
  k_dw <<<dim3(2, 768, 2),   blk, 0, stream>>>(qkv, wdw, agg);
  k_pw <<<dim3(128, 24, 2),  blk, 0, stream>>>(agg, wpw, agg2);
  k_vkp<<<dim3(16, 32, 1),   blk, 0, stream>>>(qkv, agg2, vkp);
  k_vkr<<<dim3(32, 1, 1),    blk, 0, stream>>>(vkp, vk);
  k_att<<<dim3(128, 16, 2),  blk, 0, stream>>>(qkv, agg2, vk, att);
  k_proj<<<dim3(128, 8, 2),  blk, 0, stream>>>(att, wproj, gamma, beta, rmean,
                                               rvar, out);
}